// SparseRKAN_10325101380248
// MI455X (gfx1250) — compile-verified
//
#include <hip/hip_runtime.h>
#include <hip/hip_bf16.h>
#include <math.h>

// ---------------- problem constants ----------------
#define Hdim   16384
#define Bdim   64
#define Tdim   64
#define NNZ    (1 << 20)
#define PANELS (Hdim / 16)      // 1024 panels of 16 rows
#define THSTRIDE (Tdim * Hdim)  // stride between batches in x / out

typedef float v2f __attribute__((ext_vector_type(2)));
typedef float v8f __attribute__((ext_vector_type(8)));

// ---------------- small utility kernels ----------------
__global__ void zero_u32_kernel(unsigned* __restrict__ p, int n) {
    int i = blockIdx.x * blockDim.x + threadIdx.x;
    if (i < n) p[i] = 0u;
}
__global__ void zero_f32_kernel(float* __restrict__ p, int n) {
    int i = blockIdx.x * blockDim.x + threadIdx.x;
    if (i < n) p[i] = 0.0f;
}

// Count nonzeros per 16-row panel. rows = idx[0][*]
__global__ void count_panels_kernel(const int* __restrict__ rows,
                                    unsigned* __restrict__ counts, int nnz) {
    int i = blockIdx.x * blockDim.x + threadIdx.x;
    if (i < nnz) {
        unsigned r = (unsigned)rows[i];
        atomicAdd(&counts[r >> 4], 1u);
    }
}

// Exclusive scan of ceil4(counts) over PANELS=1024 buckets (single block).
__global__ void scan_panels_kernel(const unsigned* __restrict__ counts,
                                   unsigned* __restrict__ start,
                                   unsigned* __restrict__ cursor) {
    __shared__ unsigned s[PANELS];
    int tid = threadIdx.x;
    unsigned c = (counts[tid] + 3u) & ~3u;   // round to multiple of 4 -> aligned chunks
    s[tid] = c;
    __syncthreads();
    for (int off = 1; off < PANELS; off <<= 1) {
        unsigned t = (tid >= off) ? s[tid - off] : 0u;
        __syncthreads();
        s[tid] += t;
        __syncthreads();
    }
    unsigned excl = s[tid] - c;
    start[tid]  = excl;
    cursor[tid] = excl;
}

// Scatter nonzeros into panel buckets as interleaved entries:
//   ev[pos] = { col | (localRow<<24), bits(val) }   (8 bytes per nonzero)
__global__ void scatter_panels_kernel(const int* __restrict__ rows,
                                      const int* __restrict__ cols,
                                      const float* __restrict__ vals,
                                      unsigned* __restrict__ cursor,
                                      uint2* __restrict__ ev, int nnz) {
    int i = blockIdx.x * blockDim.x + threadIdx.x;
    if (i < nnz) {
        unsigned r = (unsigned)rows[i];
        unsigned pos = atomicAdd(&cursor[r >> 4], 1u);
        ev[pos] = make_uint2(((unsigned)cols[i] & 16383u) | ((r & 15u) << 24),
                             __float_as_uint(vals[i]));
    }
}

// Zero the <=3 padding entries per panel between cnt and ceil4(cnt) so padded
// chunk slots contribute exactly 0 (col=0, val=0) -> no tail logic needed.
__global__ void pad_gaps_kernel(const unsigned* __restrict__ start,
                                const unsigned* __restrict__ counts,
                                uint2* __restrict__ ev) {
    int p = blockIdx.x * blockDim.x + threadIdx.x;
    if (p < PANELS) {
        unsigned b = start[p] + counts[p];
        unsigned e = start[p] + ((counts[p] + 3u) & ~3u);
        for (unsigned i = b; i < e; ++i) ev[i] = make_uint2(0u, 0u);
    }
}

// Transpose x[:, t, :] (B,H slice of (B,T,H)) -> xt (H,B), LDS-tiled.
__global__ void transpose_xt_kernel(const float* __restrict__ x, int t,
                                    float* __restrict__ xt) {
    __shared__ float s[64][65];
    const int h0  = blockIdx.x * 64;
    const int tid = threadIdx.x;       // 256 threads
#pragma unroll
    for (int r = 0; r < 16; ++r) {     // coalesced along h
        int b = r * 4 + (tid >> 6);
        int h = tid & 63;
        s[h][b] = x[(size_t)b * THSTRIDE + (size_t)t * Hdim + h0 + h];
    }
    __syncthreads();
#pragma unroll
    for (int r = 0; r < 16; ++r) {     // coalesced along b
        int h = r * 4 + (tid >> 6);
        int b = tid & 63;
        xt[(size_t)(h0 + h) * Bdim + b] = s[h][b];
    }
}

// ---------------- WMMA panel accumulation ----------------
// For 4 nonzeros (k=0..3) of a 16-row panel:
//   A[16x4]: A[m][k] = val_k if localRow_k == m else 0
//   B[4x16]: B[k][n] = src[col_k * 64 + q*16 + n]
//   C += A x B  accumulates val_k * src[col_k, :] into row localRow_k.
// A layout (16x4 f32): lane m   : v0=A[m][0], v1=A[m][1]
//                      lane m+16: v0=A[m][2], v1=A[m][3]
// B layout mirrors with N striped across lanes.
// `src` is uniform (SGPR base) and `bofs` is the per-lane 32-bit offset, so the
// gathers lower to saddr+voffset global_load_b32 (scale_offset form).
// The A gate uses  val * (lrow==m ? 1 : 0)  so the 16B stream load is consumed
// unconditionally (prevents the compiler predicating part of the load).
__device__ __forceinline__ void panel_accum(const uint2* __restrict__ ev,
                                            unsigned base, unsigned cnt,
                                            const float* __restrict__ src,
                                            unsigned bofs, int lane, v8f& acc) {
    const unsigned m  = (unsigned)(lane & 15);
    const int      k0 = (lane >> 4) * 2;       // this lane supplies entries k0, k0+1
    const unsigned end4 = base + ((cnt + 3u) & ~3u);   // gaps are zeroed -> no tail
#pragma unroll 2
    for (unsigned p = base; p < end4; p += 4) {
        // one 16B load: (pk0, val0, pk1, val1); aligned (base%4==0, k0 even)
        uint4 e = *(const uint4*)(ev + p + k0);
        float s0 = ((e.x >> 24) == m) ? 1.0f : 0.0f;
        float s1 = ((e.z >> 24) == m) ? 1.0f : 0.0f;
        float a0 = __uint_as_float(e.y) * s0;
        float a1 = __uint_as_float(e.w) * s1;
        float b0 = src[(e.x & 16383u) * Bdim + bofs];
        float b1 = src[(e.z & 16383u) * Bdim + bofs];
        v2f A  = {a0, a1};
        v2f Bv = {b0, b1};
        acc = __builtin_amdgcn_wmma_f32_16x16x4_f32(false, A, false, Bv,
                                                    (short)0, acc, false, false);
    }
}

// Fused recurrent step: y = tanh(W_ih @ xt + b_ih + W_hh @ hprev + b_hh)
// One block = one 16-row panel; 4 waves = 4 batch quarters.
__global__ __launch_bounds__(128)
void spmm_step_kernel(const float* __restrict__ xt, const float* __restrict__ hprev,
                      const uint2* __restrict__ ev_ih,
                      const unsigned* __restrict__ st_ih, const unsigned* __restrict__ cn_ih,
                      const uint2* __restrict__ ev_hh,
                      const unsigned* __restrict__ st_hh, const unsigned* __restrict__ cn_hh,
                      const float* __restrict__ bias_ih, const float* __restrict__ bias_hh,
                      float* __restrict__ hnext, float* __restrict__ out_t,
                      float* __restrict__ hfinal) {
    __shared__ float tile[16][65];
    const int panel = blockIdx.x;
    const int tid   = threadIdx.x;
    const int lane  = tid & 31;
    const int q     = tid >> 5;                    // batch quarter 0..3
    const unsigned bofs = (unsigned)(q * 16 + (lane & 15));

    v8f acc = {};
    panel_accum(ev_ih, st_ih[panel], cn_ih[panel], xt,    bofs, lane, acc);
    panel_accum(ev_hh, st_hh[panel], cn_hh[panel], hprev, bofs, lane, acc);

    // Epilogue: bias + tanh, stage into LDS.
    // C layout: lane L, vgpr v -> row = v + 8*(L>=16), col n = L&15
    const int hi   = lane >> 4;
    const int n    = lane & 15;
    const int row0 = panel * 16;
#pragma unroll
    for (int v = 0; v < 8; ++v) {
        int r = v + 8 * hi;
        float val = acc[v] + bias_ih[row0 + r] + bias_hh[row0 + r];
        tile[r][q * 16 + n] = tanhf(val);
    }
    __syncthreads();

    // h_next (H,B): row-contiguous 32B stores
    {
        int r  = tid >> 3;          // 0..15
        int b8 = (tid & 7) * 8;     // 0..56
        float* d = hnext + (size_t)(row0 + r) * Bdim + b8;
        *(float4*)(d + 0) = make_float4(tile[r][b8 + 0], tile[r][b8 + 1],
                                        tile[r][b8 + 2], tile[r][b8 + 3]);
        *(float4*)(d + 4) = make_float4(tile[r][b8 + 4], tile[r][b8 + 5],
                                        tile[r][b8 + 6], tile[r][b8 + 7]);
    }
    // out[b][t][row0..row0+15]: transposed through LDS, 32B runs per batch
    {
        int b  = tid >> 1;          // 0..63
        int hs = (tid & 1) * 8;     // 0 or 8
        float4 p0 = make_float4(tile[hs + 0][b], tile[hs + 1][b],
                                tile[hs + 2][b], tile[hs + 3][b]);
        float4 p1 = make_float4(tile[hs + 4][b], tile[hs + 5][b],
                                tile[hs + 6][b], tile[hs + 7][b]);
        float* d = out_t + (size_t)b * THSTRIDE + row0 + hs;
        *(float4*)(d + 0) = p0;
        *(float4*)(d + 4) = p1;
        if (hfinal) {               // last step also emits h_final (B,1,H)
            float* d2 = hfinal + (size_t)b * Hdim + row0 + hs;
            *(float4*)(d2 + 0) = p0;
            *(float4*)(d2 + 4) = p1;
        }
    }
}

// ---------------- host-side launch ----------------
extern "C" void kernel_launch(void* const* d_in, const int* in_sizes, int n_in,
                              void* d_out, int out_size, void* d_ws, size_t ws_size,
                              hipStream_t stream) {
    (void)in_sizes; (void)n_in; (void)out_size; (void)ws_size;

    const float* x       = (const float*)d_in[0];            // (B,T,H)
    const int*   idx_ih  = (const int*)d_in[1];              // (2,NNZ): rows then cols
    const float* vals_ih = (const float*)d_in[2];
    const int*   idx_hh  = (const int*)d_in[3];
    const float* vals_hh = (const float*)d_in[4];
    const float* bias_ih = (const float*)d_in[5];            // (H,1)
    const float* bias_hh = (const float*)d_in[6];

    float* out    = (float*)d_out;                           // (B,T,H)
    float* hfinal = out + (size_t)Bdim * Tdim * Hdim;        // (B,1,H)

    // ---- carve workspace ----
    char* ws = (char*)d_ws;
    auto carve = [&](size_t bytes) -> void* {
        void* p = (void*)ws;
        ws += (bytes + 255) & ~(size_t)255;
        return p;
    };
    const size_t STREAM_CAP = (size_t)NNZ + 4096;            // padded for ceil4 gaps
    unsigned* cnt_ih = (unsigned*)carve(PANELS * 4);
    unsigned* cnt_hh = (unsigned*)carve(PANELS * 4);
    unsigned* st_ih  = (unsigned*)carve(PANELS * 4);
    unsigned* st_hh  = (unsigned*)carve(PANELS * 4);
    unsigned* cur_ih = (unsigned*)carve(PANELS * 4);
    unsigned* cur_hh = (unsigned*)carve(PANELS * 4);
    uint2*    ev_ih  = (uint2*)   carve(STREAM_CAP * 8);
    uint2*    ev_hh  = (uint2*)   carve(STREAM_CAP * 8);
    float*    xt     = (float*)   carve((size_t)Hdim * Bdim * 4);
    float*    hbufA  = (float*)   carve((size_t)Hdim * Bdim * 4);
    float*    hbufB  = (float*)   carve((size_t)Hdim * Bdim * 4);

    // ---- preprocessing: COO -> panel-bucketed streams (rebuilt every call) ----
    zero_u32_kernel<<<(PANELS + 255) / 256, 256, 0, stream>>>(cnt_ih, PANELS);
    zero_u32_kernel<<<(PANELS + 255) / 256, 256, 0, stream>>>(cnt_hh, PANELS);

    const int nb = (NNZ + 255) / 256;
    count_panels_kernel<<<nb, 256, 0, stream>>>(idx_ih, cnt_ih, NNZ);
    count_panels_kernel<<<nb, 256, 0, stream>>>(idx_hh, cnt_hh, NNZ);

    scan_panels_kernel<<<1, PANELS, 0, stream>>>(cnt_ih, st_ih, cur_ih);
    scan_panels_kernel<<<1, PANELS, 0, stream>>>(cnt_hh, st_hh, cur_hh);

    scatter_panels_kernel<<<nb, 256, 0, stream>>>(idx_ih, idx_ih + NNZ, vals_ih,
                                                  cur_ih, ev_ih, NNZ);
    scatter_panels_kernel<<<nb, 256, 0, stream>>>(idx_hh, idx_hh + NNZ, vals_hh,
                                                  cur_hh, ev_hh, NNZ);

    pad_gaps_kernel<<<(PANELS + 255) / 256, 256, 0, stream>>>(st_ih, cnt_ih, ev_ih);
    pad_gaps_kernel<<<(PANELS + 255) / 256, 256, 0, stream>>>(st_hh, cnt_hh, ev_hh);

    // h0 = 0
    zero_f32_kernel<<<(Hdim * Bdim + 255) / 256, 256, 0, stream>>>(hbufA, Hdim * Bdim);

    // ---- recurrent timestep loop ----
    float* hprev = hbufA;
    float* hnext = hbufB;
    for (int t = 0; t < Tdim; ++t) {
        transpose_xt_kernel<<<Hdim / 64, 256, 0, stream>>>(x, t, xt);
        spmm_step_kernel<<<PANELS, 128, 0, stream>>>(
            xt, hprev,
            ev_ih, st_ih, cnt_ih,
            ev_hh, st_hh, cnt_hh,
            bias_ih, bias_hh,
            hnext, out + (size_t)t * Hdim,
            (t == Tdim - 1) ? hfinal : nullptr);
        float* tmp = hprev; hprev = hnext; hnext = tmp;
    }
}